// MultiHeadAttention_39092792328315
// MI455X (gfx1250) — compile-verified
//
#include <hip/hip_runtime.h>
#include <hip/hip_bf16.h>
#include <math.h>

// -----------------------------------------------------------------------------
// MHA forward on gfx1250 (CDNA5, wave32) using fp32 WMMA (v_wmma_f32_16x16x4_f32)
//   q = x@Wq+bq ; k = x@Wk+bk ; v = x@Wv+bv          (tiled LDS GEMM, WMMA f32)
//   o = softmax(causal(q k^T / 8)) v                 (flash-attn, WMMA f32)
//   out = o@Wo+bo                                    (tiled LDS GEMM, WMMA f32)
// Workspace layout (d_ws): Q | K | V | attn_out, 4 x 8192*1024 floats = 128 MB.
// -----------------------------------------------------------------------------

typedef float v2f __attribute__((ext_vector_type(2)));
typedef float v4f __attribute__((ext_vector_type(4)));
typedef float v8f __attribute__((ext_vector_type(8)));

#define WMMA_F32_16x16x4(a, b, c) \
  __builtin_amdgcn_wmma_f32_16x16x4_f32(false, (a), false, (b), (short)0, (c), false, false)

// A-matrix 16x4 layout (wave32): lane ll=lane&15 -> M row; half lh=lane>>4 picks
// K={0,1} (lh=0) or K={2,3} (lh=1); a.x=K even, a.y=K odd.
// B-matrix 4x16: b.x = B[2*lh + 0][ll], b.y = B[2*lh + 1][ll].
// C/D 16x16: c[i] = C[i + 8*lh][ll].

#define TM 256   // workgroup M tile (8 waves x 32 rows)
#define TN 64    // workgroup N tile
#define TK 16    // K chunk

// C[M,N] = A[M,K] @ W[K,N] + bias.  Each wave computes a 32x64 strip:
// 2 M-subtiles x 4 N-subtiles, so each B fragment (one ds_load_b64 thanks to
// the pair-interleaved LDS layout) feeds 2 WMMAs and each A pair feeds 4.
__global__ __launch_bounds__(256) void gemm_bias_f32(
    const float* __restrict__ A, const float* __restrict__ W,
    const float* __restrict__ bias, float* __restrict__ C,
    int M, int N, int K)
{
  __shared__ float As[TM * TK];           // [row][k], 16 KB
  __shared__ float Bp[(TK / 2) * TN * 2]; // pair-interleaved [k/2][n][k&1], 4 KB

  const int tid  = threadIdx.x;
  const int lane = tid & 31;
  const int wave = tid >> 5;        // 0..7
  const int ll = lane & 15;
  const int lh = lane >> 4;
  const int bm = blockIdx.x * TM;
  const int bn = blockIdx.y * TN;

  v8f acc[2][4];
  for (int mt = 0; mt < 2; ++mt)
    for (int nt = 0; nt < 4; ++nt)
      for (int i = 0; i < 8; ++i) acc[mt][nt][i] = 0.0f;

  for (int k0 = 0; k0 < K; k0 += TK) {
    __syncthreads();
    // ---- stage A tile: 256x16 floats = 1024 float4, 4 per thread
    {
      int idx = tid;
      for (int r = 0; r < 4; ++r, idx += 256) {
        const int row = idx >> 2;              // 0..255
        const int c4  = (idx & 3) << 2;        // 0,4,8,12
        v4f val = *(const v4f*)(A + (size_t)(bm + row) * K + (k0 + c4));
        *(v4f*)(As + row * TK + c4) = val;
      }
      // ---- stage B tile: 16x64 floats, pair-interleave scatter (1 float4/thread)
      const int row = tid >> 4;                // k: 0..15
      const int c4  = (tid & 15) << 2;         // n: 0..60
      v4f val = *(const v4f*)(W + (size_t)(k0 + row) * N + (bn + c4));
      float* bdst = Bp + (row >> 1) * (TN * 2) + (row & 1);
      bdst[(c4 + 0) * 2] = val.x;
      bdst[(c4 + 1) * 2] = val.y;
      bdst[(c4 + 2) * 2] = val.z;
      bdst[(c4 + 3) * 2] = val.w;
      // hint next K-chunk toward L2 (global_prefetch_b8)
      if (k0 + TK < K) {
        __builtin_prefetch(A + (size_t)(bm + tid) * K + (k0 + TK), 0, 1);
        __builtin_prefetch(W + (size_t)(k0 + TK + row) * N + (bn + c4), 0, 1);
      }
    }
    __syncthreads();

    const float* a0 = As + (wave * 32 + ll) * TK;   // M-subtile 0 row
    const float* a1 = a0 + 16 * TK;                 // M-subtile 1 row
    for (int kk = 0; kk < 4; ++kk) {
      const int kb = kk * 4 + 2 * lh;
      v2f af0 = *(const v2f*)(a0 + kb);
      v2f af1 = *(const v2f*)(a1 + kb);
      const float* brow = Bp + (kk * 2 + lh) * (TN * 2);   // (kb>>1) pair row
      for (int nt = 0; nt < 4; ++nt) {
        v2f bf = *(const v2f*)(brow + (nt * 16 + ll) * 2); // one ds_load_b64
        acc[0][nt] = WMMA_F32_16x16x4(af0, bf, acc[0][nt]);
        acc[1][nt] = WMMA_F32_16x16x4(af1, bf, acc[1][nt]);
      }
    }
  }

  // epilogue: + bias, store
  for (int mt = 0; mt < 2; ++mt) {
    for (int nt = 0; nt < 4; ++nt) {
      const int col = bn + nt * 16 + ll;
      const float bv = bias[col];
      for (int i = 0; i < 8; ++i) {
        const int row = bm + wave * 32 + mt * 16 + i + 8 * lh;
        C[(size_t)row * N + col] = acc[mt][nt][i] + bv;
      }
    }
  }
}

// -----------------------------------------------------------------------------
// Flash attention with causal mask, fp32 WMMA. One wave = one 16-query tile.
// Block = 128 threads (4 waves, 4 adjacent q-tiles of the same (b,h)).
// -----------------------------------------------------------------------------
__global__ __launch_bounds__(128) void attn_causal_f32(
    const float* __restrict__ Qg, const float* __restrict__ Kg,
    const float* __restrict__ Vg, float* __restrict__ Og,
    int B, int T, int H)
{
  __shared__ float ptile[4][16 * 16];    // per-wave P tile (C-layout -> A-layout)

  const int lane = threadIdx.x & 31;
  const int wid  = threadIdx.x >> 5;
  const int ll = lane & 15;
  const int lh = lane >> 4;
  const int D = H * 64;

  const int tilesPerBH  = T / 16;                 // 256
  const int blocksPerBH = tilesPerBH / 4;         // 64
  const int bh = blockIdx.x / blocksPerBH;
  const int b  = bh / H;
  const int h  = bh % H;
  const int q0 = ((blockIdx.x % blocksPerBH) * 4 + wid) * 16;

  const float* Qh = Qg + (size_t)b * T * D + h * 64;
  const float* Kh = Kg + (size_t)b * T * D + h * 64;
  const float* Vh = Vg + (size_t)b * T * D + h * 64;

  // preload Q fragments: 16 K-steps covering d = 0..63
  v2f qf[16];
  {
    const float* qrow = Qh + (size_t)(q0 + ll) * D;
    for (int ds = 0; ds < 16; ++ds)
      qf[ds] = *(const v2f*)(qrow + ds * 4 + 2 * lh);
  }

  v8f oacc[4];
  for (int nt = 0; nt < 4; ++nt)
    for (int i = 0; i < 8; ++i) oacc[nt][i] = 0.0f;
  float mrow[8], lrow[8];
  for (int i = 0; i < 8; ++i) { mrow[i] = -3.0e38f; lrow[i] = 0.0f; }

  const float scale = 0.125f;  // 1/sqrt(64)

  for (int kb = 0; kb <= q0; kb += 16) {         // causal: key blocks <= q-tile
    // ---- S = Q K^T (16x16), K-dim = 64 in 16 WMMA steps
    v8f s;
    for (int i = 0; i < 8; ++i) s[i] = 0.0f;
    {
      const float* krow = Kh + (size_t)(kb + ll) * D;
      for (int ds = 0; ds < 16; ++ds) {
        v2f bf = *(const v2f*)(krow + ds * 4 + 2 * lh);  // K^T[d][key=ll]
        s = WMMA_F32_16x16x4(qf[ds], bf, s);
      }
    }
    // ---- scale + causal mask (branch is wave-uniform; mask per-lane select)
    if (kb == q0) {
      for (int i = 0; i < 8; ++i) {
        const int qr = i + 8 * lh;
        s[i] = (ll <= qr) ? s[i] * scale : -3.0e38f;
      }
    } else {
      for (int i = 0; i < 8; ++i) s[i] *= scale;
    }
    // ---- online softmax: row reductions across 16-lane halves
    float alpha[8];
    for (int i = 0; i < 8; ++i) {
      float mx = s[i];
      mx = fmaxf(mx, __shfl_xor(mx, 1, 32));
      mx = fmaxf(mx, __shfl_xor(mx, 2, 32));
      mx = fmaxf(mx, __shfl_xor(mx, 4, 32));
      mx = fmaxf(mx, __shfl_xor(mx, 8, 32));
      const float mnew = fmaxf(mrow[i], mx);
      const float a = __expf(mrow[i] - mnew);
      const float p = __expf(s[i] - mnew);
      float ps = p;
      ps += __shfl_xor(ps, 1, 32);
      ps += __shfl_xor(ps, 2, 32);
      ps += __shfl_xor(ps, 4, 32);
      ps += __shfl_xor(ps, 8, 32);
      lrow[i] = lrow[i] * a + ps;
      mrow[i] = mnew;
      alpha[i] = a;
      ptile[wid][(i + 8 * lh) * 16 + ll] = p;    // C-layout -> LDS [q][k]
    }
    for (int nt = 0; nt < 4; ++nt)
      for (int i = 0; i < 8; ++i) oacc[nt][i] *= alpha[i];

    // ---- O += P @ V (16x16 x 16x64), A-frag of P re-read from LDS
    for (int kk = 0; kk < 4; ++kk) {
      const int kr = kk * 4 + 2 * lh;
      v2f p = *(const v2f*)(&ptile[wid][ll * 16 + kr]);   // P[q=ll][kr], [kr+1]
      const float* vr0 = Vh + (size_t)(kb + kr) * D;
      const float* vr1 = vr0 + D;
      for (int nt = 0; nt < 4; ++nt) {
        v2f bf;
        bf.x = vr0[nt * 16 + ll];
        bf.y = vr1[nt * 16 + ll];
        oacc[nt] = WMMA_F32_16x16x4(p, bf, oacc[nt]);
      }
    }
  }

  // ---- normalize and store to [B,T,D] (head-interleaved) layout
  for (int i = 0; i < 8; ++i) {
    const float inv = 1.0f / lrow[i];
    const int qr = q0 + i + 8 * lh;
    float* orow = Og + (size_t)(b * T + qr) * D + h * 64;
    for (int nt = 0; nt < 4; ++nt)
      orow[nt * 16 + ll] = oacc[nt][i] * inv;
  }
}

// -----------------------------------------------------------------------------
extern "C" void kernel_launch(void* const* d_in, const int* in_sizes, int n_in,
                              void* d_out, int out_size, void* d_ws, size_t ws_size,
                              hipStream_t stream)
{
  const float* x  = (const float*)d_in[0];
  const float* Wq = (const float*)d_in[1];
  const float* bq = (const float*)d_in[2];
  const float* Wk = (const float*)d_in[3];
  const float* bk = (const float*)d_in[4];
  const float* Wv = (const float*)d_in[5];
  const float* bv = (const float*)d_in[6];
  const float* Wo = (const float*)d_in[7];
  const float* bo = (const float*)d_in[8];
  float* out = (float*)d_out;

  const int B = 2, T = 4096, D = 1024, H = 16;
  const int M = B * T;                          // 8192
  const size_t sz = (size_t)M * D;              // 8M floats

  float* Qb = (float*)d_ws;                     // needs 4*sz*4 = 128 MB
  float* Kb = Qb + sz;
  float* Vb = Kb + sz;
  float* Ab = Vb + sz;

  dim3 gBlk(256);
  dim3 gGrd(M / TM, D / TN);                    // 32 x 16

  hipLaunchKernelGGL(gemm_bias_f32, gGrd, gBlk, 0, stream, x, Wq, bq, Qb, M, D, D);
  hipLaunchKernelGGL(gemm_bias_f32, gGrd, gBlk, 0, stream, x, Wk, bk, Kb, M, D, D);
  hipLaunchKernelGGL(gemm_bias_f32, gGrd, gBlk, 0, stream, x, Wv, bv, Vb, M, D, D);

  dim3 aBlk(128);
  dim3 aGrd(B * H * (T / 16) / 4);              // 2048 blocks, 4 q-tiles each
  hipLaunchKernelGGL(attn_causal_f32, aGrd, aBlk, 0, stream, Qb, Kb, Vb, Ab, B, T, H);

  hipLaunchKernelGGL(gemm_bias_f32, gGrd, gBlk, 0, stream, Ab, Wo, bo, out, M, D, D);
}